// GoConv3D_5626407158177
// MI455X (gfx1250) — compile-verified
//
#include <hip/hip_runtime.h>
#include <math.h>

// Problem constants (from reference setup_inputs)
#define Cc     8
#define Dd     32
#define Hh     128
#define Ww     128
#define K2     9
#define OC     16
#define KSUM   216          // (Cc*K2)*3 depth taps
#define HW     (Hh*Ww)      // 16384
#define NTOT   (Dd*Hh*Ww)   // 524288 positions per out channel
#define NT_F   524288.0f
#define WAVES_PER_BLK 4
#define TILES  (Dd*Hh*(Ww/16))   // 32768 N-tiles of 16

typedef float v2f __attribute__((ext_vector_type(2)));
typedef float v8f __attribute__((ext_vector_type(8)));

__global__ void zero_stats_kernel(float* stats) {
    if (threadIdx.x < 32) stats[threadIdx.x] = 0.0f;
}

// Fused: gca_cost (computed on the fly into LDS) -> conv3d-as-GEMM via
// V_WMMA_F32_16X16X4_F32 -> y staging + per-channel sum/sumsq reduction.
__global__ __launch_bounds__(WAVES_PER_BLK * 32)
void gca_conv_wmma_kernel(const float* __restrict__ x,
                          const float* __restrict__ dp,
                          const float* __restrict__ normal,
                          const float* __restrict__ intri,
                          const float* __restrict__ weight,
                          float* __restrict__ y,
                          float* __restrict__ gstats) {
    // Per-wave cost panel: 216 K-rows x 16 N-cols (fp32). 13.5KB/wave.
    __shared__ float costL[WAVES_PER_BLK][KSUM * 16];
    __shared__ float blkSum[OC];
    __shared__ float blkSq[OC];

    const int lane = threadIdx.x & 31;
    const int wv   = threadIdx.x >> 5;

    if (threadIdx.x < OC) { blkSum[threadIdx.x] = 0.0f; blkSq[threadIdx.x] = 0.0f; }

    // ---- decode this wave's 16-wide N tile: (dd, i, j0..j0+15) ----
    const int tile = blockIdx.x * WAVES_PER_BLK + wv;   // < 32768
    const int dd   = tile >> 10;                        // 128*8 tiles per depth slice
    const int rem  = tile & 1023;
    const int i    = rem >> 3;
    const int j0   = (rem & 7) << 4;

    const float fxv = intri[0], cxv = intri[2];
    const float fyv = intri[4], cyv = intri[5];

    // ---- Phase 1: fill cost panel. 9 kk * 3 kd * 16 n = 432 slots, 8 ci each ----
    for (int t = lane; t < K2 * 3 * 16; t += 32) {
        const int n  = t & 15;
        const int r2 = t >> 4;          // 0..26
        const int kd = r2 % 3;
        const int kk = r2 / 3;
        const int j  = j0 + n;
        const int ddp = dd + kd - 1;    // conv depth tap (zero-pad outside)

        if (ddp < 0 || ddp >= Dd) {
            #pragma unroll
            for (int ci = 0; ci < Cc; ++ci)
                costL[wv][((ci * K2 + kk) * 3 + kd) * 16 + n] = 0.0f;
            continue;
        }
        const int di = kk / 3, dj = kk % 3;
        int pi = i + di - 1; pi = pi < 0 ? 0 : (pi > Hh - 1 ? Hh - 1 : pi);
        int pj = j + dj - 1; pj = pj < 0 ? 0 : (pj > Ww - 1 ? Ww - 1 : pj);
        const int nb = pi * Ww + pj;

        // plane-induced depth weight
        const float pu  = ((float)pj - cxv) / fxv;
        const float pv  = ((float)pi - cyv) / fyv;
        const float puc = ((float)j  - cxv) / fxv;
        const float pvc = ((float)i  - cyv) / fyv;
        const float nx = normal[nb];
        const float ny = normal[HW + nb];
        const float nz = normal[2 * HW + nb];
        const float num = nx * puc + ny * pvc + nz;
        float den = nx * pu + ny * pv + nz;
        if (fabsf(den) < 1e-8f)
            den = (den > 0.0f) ? 1e-8f : ((den < 0.0f) ? -1e-8f : 0.0f);
        float wgt = num / den;
        if (!__builtin_isfinite(wgt)) wgt = 1.0f;

        // fractional depth coordinate at the neighbor pixel
        const float dpc = dp[ddp * HW + i * Ww + j];       // center-pixel plane depth
        const float dp0 = dp[nb];
        const float itv = dp[HW + nb] - dp0;
        const float z   = (wgt * dpc - dp0) / itv;
        const float z0f = floorf(z);
        const int   z0  = (int)z0f;
        const float fz  = z - z0f;
        const int zi0 = z0     < 0 ? 0 : (z0     > Dd - 1 ? Dd - 1 : z0);
        const int zi1 = z0 + 1 < 0 ? 0 : (z0 + 1 > Dd - 1 ? Dd - 1 : z0 + 1);
        const float a0 = (z0     >= 0 && z0     < Dd) ? (1.0f - fz) : 0.0f;
        const float a1 = (z0 + 1 >= 0 && z0 + 1 < Dd) ? fz          : 0.0f;

        const float* xp = x + nb;                 // + (ci*Dd + zi)*HW
        #pragma unroll
        for (int ci = 0; ci < Cc; ++ci) {
            const float x0 = xp[(size_t)(ci * Dd + zi0) * HW];
            const float x1 = xp[(size_t)(ci * Dd + zi1) * HW];
            costL[wv][((ci * K2 + kk) * 3 + kd) * 16 + n] = a0 * x0 + a1 * x1;
        }
    }
    __syncthreads();

    // ---- Phase 2: GEMM 16x16 tile, K = 216 in 54 steps of V_WMMA_F32_16X16X4_F32 ----
    // A (16x4 f32): lanes 0-15 hold M=0..15 {K0,K1}, lanes 16-31 hold {K2,K3}.
    // B (4x16 f32): mirrored; lane%16 = N, lane half selects K pair.
    const int m    = lane & 15;        // M index (out channel low)
    const int n    = lane & 15;        // N index within tile
    const int half = lane >> 4;
    const float* wrow = weight + m * KSUM;

    v8f acc = {};
    for (int g = 0; g < KSUM / 4; ++g) {
        const int kb = 4 * g + 2 * half;
        v2f A, B;
        A.x = wrow[kb];
        A.y = wrow[kb + 1];
        B.x = costL[wv][(kb    ) * 16 + n];
        B.y = costL[wv][(kb + 1) * 16 + n];
        acc = __builtin_amdgcn_wmma_f32_16x16x4_f32(
            /*neg_a=*/false, A, /*neg_b=*/false, B,
            /*c_mod=*/(short)0, acc, /*reuse_a=*/false, /*reuse_b=*/false);
    }

    // ---- Phase 3: stage y, reduce BN stats ----
    // C/D layout: vgpr r, lanes 0-15 -> M=r ; lanes 16-31 -> M=r+8 ; N=lane%16.
    const size_t posBase = ((size_t)(dd * Hh + i)) * Ww + j0;
    #pragma unroll
    for (int r = 0; r < 8; ++r) {
        const int o = r + 8 * half;
        y[(size_t)o * NTOT + posBase + n] = acc[r];
    }
    #pragma unroll
    for (int r = 0; r < 8; ++r) {
        float s = acc[r];
        float q = s * s;
        #pragma unroll
        for (int msk = 1; msk < 16; msk <<= 1) {
            s += __shfl_xor(s, msk, 32);
            q += __shfl_xor(q, msk, 32);
        }
        if ((lane & 15) == 0) {
            atomicAdd(&blkSum[r + 8 * half], s);
            atomicAdd(&blkSq [r + 8 * half], q);
        }
    }
    __syncthreads();
    if (threadIdx.x < OC) {
        atomicAdd(&gstats[threadIdx.x],       blkSum[threadIdx.x]);
        atomicAdd(&gstats[OC + threadIdx.x],  blkSq [threadIdx.x]);
    }
}

__global__ void bn_relu_kernel(const float* __restrict__ y,
                               const float* __restrict__ gstats,
                               const float* __restrict__ gamma,
                               const float* __restrict__ beta,
                               float* __restrict__ out) {
    const size_t idx = (size_t)blockIdx.x * blockDim.x + threadIdx.x;
    if (idx >= (size_t)OC * NTOT) return;
    const int o = (int)(idx >> 19);           // NTOT == 2^19
    const float mean = gstats[o] * (1.0f / NT_F);
    const float var  = gstats[OC + o] * (1.0f / NT_F) - mean * mean;
    const float v = gamma[o] * (y[idx] - mean) * rsqrtf(var + 1e-5f) + beta[o];
    out[idx] = v > 0.0f ? v : 0.0f;
}

extern "C" void kernel_launch(void* const* d_in, const int* in_sizes, int n_in,
                              void* d_out, int out_size, void* d_ws, size_t ws_size,
                              hipStream_t stream) {
    const float* x      = (const float*)d_in[0];
    const float* dp     = (const float*)d_in[1];
    const float* normal = (const float*)d_in[2];
    const float* intri  = (const float*)d_in[3];
    const float* weight = (const float*)d_in[4];
    const float* gamma  = (const float*)d_in[5];
    const float* beta   = (const float*)d_in[6];

    float* yb    = (float*)d_ws;                                   // 16*524288 f32 = 32 MiB
    float* stats = (float*)((char*)d_ws + (size_t)OC * NTOT * 4);  // sums[16] + sumsq[16]

    zero_stats_kernel<<<1, 32, 0, stream>>>(stats);
    gca_conv_wmma_kernel<<<TILES / WAVES_PER_BLK, WAVES_PER_BLK * 32, 0, stream>>>(
        x, dp, normal, intri, weight, yb, stats);
    const size_t total = (size_t)OC * NTOT;
    bn_relu_kernel<<<(int)((total + 255) / 256), 256, 0, stream>>>(
        yb, stats, gamma, beta, (float*)d_out);
}